// Head_75118978007668
// MI455X (gfx1250) — compile-verified
//
#include <hip/hip_runtime.h>
#include <stdint.h>

typedef __attribute__((ext_vector_type(2))) float v2f;
typedef __attribute__((ext_vector_type(8))) float v8f;

#define S_LEN 2048
#define D_DIM 128
#define B_DIM 4
#define NTOK (B_DIM * S_LEN)     // 8192 tokens (M)
#define K_TOT (D_DIM * D_DIM)    // 16384 (K)
#define XP 132                   // padded stride for x/s tiles (floats)
#define WKK 16                   // K elements per streamed W chunk
#define WSTR 136                 // padded floats per kk-row of W chunk
                                 // (2 rows = 272 = 16 mod 64 banks -> the two
                                 //  half-waves hit disjoint bank groups)
#define NCHUNK (K_TOT / WKK)     // 1024 chunks

// ---------------------------------------------------------------------------
// CDNA5 async global->LDS copy (ASYNCcnt path).  Writes LDS directly from the
// memory system; no VGPR data staging.  Generic LDS pointer's low 32 bits are
// the LDS offset (ISA 10.2 aperture rule).
// ---------------------------------------------------------------------------
__device__ __forceinline__ void async_copy16(float* lds_dst, const float* gsrc)
{
#if defined(__gfx1250__)
    asm volatile("global_load_async_to_lds_b128 %0, %1, off"
                 :
                 : "v"((uint32_t)(uintptr_t)lds_dst), "v"(gsrc)
                 : "memory");
#else
    *(float4*)lds_dst = *(const float4*)gsrc;  // host-pass / other targets
#endif
}

__device__ __forceinline__ void wait_async_zero()
{
#if defined(__gfx1250__)
    asm volatile("s_wait_asynccnt 0x0" ::: "memory");
#endif
}

// ---------------------------------------------------------------------------
// Kernel 1: decayed prefix scan  s_t = (s_{t-1} + x_{t-1}) / 1.2,  s_0 = 0
// ---------------------------------------------------------------------------
__global__ __launch_bounds__(D_DIM) void scan_kernel(const float* __restrict__ x,
                                                     float* __restrict__ s)
{
    const int b = blockIdx.x;
    const int d = threadIdx.x;
    const float inv = 1.0f / 1.2f;
    const float* xp = x + (size_t)b * S_LEN * D_DIM + d;
    float* sp = s + (size_t)b * S_LEN * D_DIM + d;
    float c = 0.0f;
#pragma unroll 8
    for (int t = 0; t < S_LEN; ++t) {
        sp[(size_t)t * D_DIM] = c;
        c = (c + xp[(size_t)t * D_DIM]) * inv;
    }
}

// ---------------------------------------------------------------------------
// Kernel 2: transpose concept_map[f, k] (k = d*128+e) -> Wt[k, f]
// ---------------------------------------------------------------------------
__global__ __launch_bounds__(256) void transpose_kernel(const float* __restrict__ cm,
                                                        float* __restrict__ wt)
{
    __shared__ float tile[32][33];
    const int k0 = blockIdx.x * 32;
    const int f0 = blockIdx.y * 32;
    const int tx = threadIdx.x & 31;
    const int ty = threadIdx.x >> 5;   // 0..7
#pragma unroll
    for (int j = 0; j < 4; ++j)
        tile[ty + j * 8][tx] = cm[(size_t)(f0 + ty + j * 8) * K_TOT + k0 + tx];
    __syncthreads();
#pragma unroll
    for (int j = 0; j < 4; ++j)
        wt[(size_t)(k0 + ty + j * 8) * D_DIM + f0 + tx] = tile[tx][ty + j * 8];
}

// ---------------------------------------------------------------------------
// Kernel 3: out = Z @ Wt,  Z[t,k] = x[t,d(k)]*s[t,e(k)]  (never materialized)
// M=8192, K=16384, N=128.  fp32 WMMA 16x16x4; W streams via double-buffered
// LDS chunks filled by async global->LDS; A fragments built with one packed
// mul from x/s tiles in LDS.
// Block: 256 threads (8 waves) = 2 M-tiles (32 tokens) x full N (128).
// ---------------------------------------------------------------------------
__global__ __launch_bounds__(256) void cmap_gemm_kernel(const float* __restrict__ x,
                                                        const float* __restrict__ s,
                                                        const float* __restrict__ wt,
                                                        float* __restrict__ out)
{
    __shared__ float xs[32 * XP];              // 16896 B
    __shared__ float ss[32 * XP];              // 16896 B
    __shared__ float wl[2][WKK * WSTR];        // 17408 B  (total 51200 B)

    const int tid = threadIdx.x;
    const int tok0 = blockIdx.x * 32;

    // Stage x/s tiles: 32 tokens x 128 channels, vectorized, padded stride.
#pragma unroll
    for (int r = 0; r < 4; ++r) {
        const int idx = (tid + r * 256) * 4;   // 0..4095
        const int m = idx >> 7;
        const int dd = idx & 127;
        const float4 vx = *(const float4*)(x + (size_t)(tok0 + m) * D_DIM + dd);
        const float4 vs = *(const float4*)(s + (size_t)(tok0 + m) * D_DIM + dd);
        *(float4*)(xs + m * XP + dd) = vx;     // (m*132+dd)%4==0 -> 16B aligned
        *(float4*)(ss + m * XP + dd) = vs;
    }

    // Per-thread W-chunk copy coordinates (2 x 16B per thread per chunk).
    const int cidx0 = tid * 4;                 // element index in 2048-elem chunk
    const int cidx1 = (tid + 256) * 4;
    const int ldst0 = (cidx0 >> 7) * WSTR + (cidx0 & 127);
    const int ldst1 = (cidx1 >> 7) * WSTR + (cidx1 & 127);

    // Fill chunk 0 asynchronously.
    async_copy16(&wl[0][ldst0], wt + cidx0);
    async_copy16(&wl[0][ldst1], wt + cidx1);
    wait_async_zero();
    __syncthreads();

    const int lane = tid & 31;
    const int wave = tid >> 5;
    const int mrow = wave >> 2;                // 0..1 : which 16-token row
    const int npair = wave & 3;                // 0..3 : which pair of N tiles
    const int f0a = npair * 32;
    const int f0b = f0a + 16;
    const int lm = lane & 15;
    const int half = lane >> 4;                // 0: K+0, 1: K+2 (A/B layout)
    const int moff = (mrow * 16 + lm) * XP;
    const int wofsA = (2 * half) * WSTR + (f0a + lm);   // float offsets into chunk
    const int wofsB = (2 * half) * WSTR + (f0b + lm);

    v8f acc0 = {};
    v8f acc1 = {};
    int p = 0;
    for (int c = 0; c < NCHUNK; ++c) {
        // Issue async fill of chunk c+1 into the other buffer (previous
        // barrier guarantees nobody still reads it).
        if (c + 1 < NCHUNK) {
            const float* src = wt + (size_t)(c + 1) * (WKK * D_DIM);
            float* dq = &wl[p ^ 1][0];
            async_copy16(dq + ldst0, src + cidx0);
            async_copy16(dq + ldst1, src + cidx1);
        }

        // Compute on wl[p]: chunk covers k = c*16 .. c*16+15  (d = c>>3 fixed)
        const int d = c >> 3;
        const int e0 = (c & 7) * WKK;
        const float xv = xs[moff + d];         // x broadcast, once per chunk
        const float* wp = &wl[p][0];
#pragma unroll
        for (int j = 0; j < 4; ++j) {
            // A fragment 16x4 f32: VGPR r, lane L -> M=L%16, K=r+2*(L/16)
            const v2f sv = *(const v2f*)(ss + moff + e0 + j * 4 + 2 * half);
            v2f a;
            a.x = xv * sv.x;
            a.y = xv * sv.y;
            // B fragment 4x16 f32: VGPR r, lane L -> N=L%16, K=r+2*(L/16)
            v2f b0, b1;
            b0.x = wp[wofsA + j * 4 * WSTR];
            b0.y = wp[wofsA + j * 4 * WSTR + WSTR];
            b1.x = wp[wofsB + j * 4 * WSTR];
            b1.y = wp[wofsB + j * 4 * WSTR + WSTR];
            acc0 = __builtin_amdgcn_wmma_f32_16x16x4_f32(false, a, false, b0,
                                                         (short)0, acc0, false, false);
            acc1 = __builtin_amdgcn_wmma_f32_16x16x4_f32(false, a, false, b1,
                                                         (short)0, acc1, false, false);
        }

        wait_async_zero();       // own async stores to wl[p^1] landed in LDS
        __syncthreads();         // everyone's landed; done reading wl[p]
        p ^= 1;
    }

    // Epilogue: C/D layout VGPR r, lane L -> M = r + 8*(L/16), N = L%16
    const int tokm = tok0 + mrow * 16;
#pragma unroll
    for (int r = 0; r < 8; ++r) {
        const int M = r + 8 * half;
        out[(size_t)(tokm + M) * D_DIM + f0a + lm] = acc0[r];
        out[(size_t)(tokm + M) * D_DIM + f0b + lm] = acc1[r];
    }
}

// ---------------------------------------------------------------------------
extern "C" void kernel_launch(void* const* d_in, const int* in_sizes, int n_in,
                              void* d_out, int out_size, void* d_ws, size_t ws_size,
                              hipStream_t stream)
{
    (void)in_sizes; (void)n_in; (void)out_size; (void)ws_size;
    const float* x  = (const float*)d_in[0];   // [4, 2048, 128] f32
    const float* cm = (const float*)d_in[1];   // [128, 128, 128] f32
    float* out = (float*)d_out;                // [4, 2048, 128] f32

    float* sbuf  = (float*)d_ws;                       // 4 MB: decayed prefix sums
    float* wtbuf = sbuf + (size_t)NTOK * D_DIM;        // 8 MB: Wt[k, f]

    scan_kernel<<<dim3(B_DIM), dim3(D_DIM), 0, stream>>>(x, sbuf);
    transpose_kernel<<<dim3(K_TOT / 32, D_DIM / 32), dim3(256), 0, stream>>>(cm, wtbuf);
    cmap_gemm_kernel<<<dim3(NTOK / 32), dim3(256), 0, stream>>>(x, sbuf, wtbuf, out);
}